// GCN_32109175505051
// MI455X (gfx1250) — compile-verified
//
#include <hip/hip_runtime.h>

// GCN (2-layer GraphConv) for MI455X / gfx1250.
// Roofline: ~2.5 GFLOP of GEMM vs ~1.4 GB of edge-scatter HBM traffic ->
// memory-bound on aggregation. GEMM runs on the matrix pipe in full fp32
// via V_WMMA_F32_16X16X4_F32 (exact reference precision).

#define GCN_N_NODES 50000
#define GCN_N_EDGES 800000
#define GCN_IN_DIM 128
#define GCN_HID_DIM 128
#define GCN_OUT_DIM 64

typedef __attribute__((ext_vector_type(2))) float v2f;
typedef __attribute__((ext_vector_type(8))) float v8f;

// ---------------------------------------------------------------------------
// Degrees: one float atomicAdd per edge endpoint (global_atomic_add_f32).
// Counts <= E = 8e5 are exactly representable in f32.
// ---------------------------------------------------------------------------
__global__ void gcn_degree_kernel(const int* __restrict__ src,
                                  const int* __restrict__ dst,
                                  float* __restrict__ deg_out,
                                  float* __restrict__ deg_in, int E) {
  int e = blockIdx.x * blockDim.x + threadIdx.x;
  if (e < E) {
    atomicAdd(&deg_out[src[e]], 1.0f);
    atomicAdd(&deg_in[dst[e]], 1.0f);
  }
}

__global__ void gcn_norm_kernel(const float* __restrict__ deg_out,
                                const float* __restrict__ deg_in,
                                float* __restrict__ norm_out,
                                float* __restrict__ norm_in, int N) {
  int i = blockIdx.x * blockDim.x + threadIdx.x;
  if (i < N) {
    norm_out[i] = rsqrtf(fmaxf(deg_out[i], 1.0f));
    norm_in[i]  = rsqrtf(fmaxf(deg_in[i], 1.0f));
  }
}

// ---------------------------------------------------------------------------
// Y[16 x NOUT] = (X[16 x K] * scale) @ W[K x NOUT], fp32 WMMA 16x16x4.
// One block = one 16-row node tile; wave w owns output columns [16w,16w+16).
// A-frag (16x4 f32): lane m = lane%16 is row M; kh = lane/16 selects K-half:
//   VGPR0 = A[m][k+2*kh], VGPR1 = A[m][k+2*kh+1].
// B-frag (4x16 f32): n = lane%16 is column; VGPR0 = B[k+2*kh][n], VGPR1 = row+1.
// C/D (16x16 f32, 8 VGPRs): row = v + 8*kh, col = lane%16.
// ---------------------------------------------------------------------------
template <int K, int NOUT>
__global__ void gcn_gemm_scaled_wmma(const float* __restrict__ X,
                                     const float* __restrict__ scale,
                                     const float* __restrict__ W,
                                     float* __restrict__ Y) {
  constexpr int LDSK = K + 4;  // pad: stride 132 floats -> banks 4*m, conflict-free
  __shared__ float tile[16 * LDSK];

  const int tid  = threadIdx.x;
  const int row0 = blockIdx.x * 16;

  // Cooperative load of the 16 x K node tile, fused with D_out^{-1/2} scaling.
  for (int t = tid; t < 16 * K; t += blockDim.x) {
    const int r = t / K;
    const int c = t - r * K;
    const int node = row0 + r;
    tile[r * LDSK + c] = X[(size_t)node * K + c] * scale[node];
  }
  __syncthreads();

  const int wave = tid >> 5;
  const int lane = tid & 31;
  const int m    = lane & 15;   // A row / B,C column-within-tile
  const int kh   = lane >> 4;   // K-half (A/B) and M-half (C/D)
  const int nc   = wave * 16 + m;

  v8f acc = {};
#pragma unroll 4
  for (int kk = 0; kk < K; kk += 4) {
    v2f a, b;
    const int ka = kk + 2 * kh;
    a.x = tile[m * LDSK + ka];
    a.y = tile[m * LDSK + ka + 1];
    b.x = W[(size_t)ka * NOUT + nc];
    b.y = W[(size_t)(ka + 1) * NOUT + nc];
    acc = __builtin_amdgcn_wmma_f32_16x16x4_f32(
        /*neg_a=*/false, a, /*neg_b=*/false, b,
        /*c_mod=*/(short)0, acc, /*reuse_a=*/false, /*reuse_b=*/false);
  }

#pragma unroll
  for (int v = 0; v < 8; ++v) {
    const int orow = row0 + v + 8 * kh;
    Y[(size_t)orow * NOUT + nc] = acc[v];
  }
}

// ---------------------------------------------------------------------------
// Edge aggregation: thread = (edge, dim). Coalesced gather of H[src] rows,
// atomic scatter-add into agg[dst] rows. This is the HBM-bound hot loop.
// ---------------------------------------------------------------------------
template <int D>
__global__ void gcn_aggregate_kernel(const float* __restrict__ H,
                                     const int* __restrict__ src,
                                     const int* __restrict__ dst,
                                     float* __restrict__ agg, int E) {
  const unsigned t = blockIdx.x * blockDim.x + threadIdx.x;
  const unsigned e = t / D;
  const unsigned d = t - e * D;
  if (e < (unsigned)E) {
    const int s = src[e];
    const int r = dst[e];
    atomicAdd(&agg[(size_t)r * D + d], H[(size_t)s * D + d]);
  }
}

// out[i,d] = relu(agg[i,d] * norm_in[i] + b[d])   (may run in-place)
template <int D>
__global__ void gcn_finalize_kernel(const float* __restrict__ agg,
                                    const float* __restrict__ norm_in,
                                    const float* __restrict__ b,
                                    float* __restrict__ out, int N) {
  const unsigned t = blockIdx.x * blockDim.x + threadIdx.x;
  const unsigned i = t / D;
  const unsigned d = t - i * D;
  if (i < (unsigned)N) {
    const float v = fmaf(agg[t], norm_in[i], b[d]);
    out[t] = fmaxf(v, 0.0f);
  }
}

extern "C" void kernel_launch(void* const* d_in, const int* in_sizes, int n_in,
                              void* d_out, int out_size, void* d_ws, size_t ws_size,
                              hipStream_t stream) {
  const float* feat = (const float*)d_in[0];
  const int*   src  = (const int*)d_in[1];
  const int*   dst  = (const int*)d_in[2];
  const float* W1   = (const float*)d_in[3];
  const float* b1   = (const float*)d_in[4];
  const float* W2   = (const float*)d_in[5];
  const float* b2   = (const float*)d_in[6];
  float* out = (float*)d_out;
  float* ws  = (float*)d_ws;

  const int N = GCN_N_NODES;
  const int E = GCN_N_EDGES;

  // Workspace layout (floats): 4N norms + 2*N*128 buffers  (~52 MB total)
  float* deg_out  = ws;
  float* deg_in   = ws + (size_t)N;
  float* norm_out = ws + 2 * (size_t)N;
  float* norm_in  = ws + 3 * (size_t)N;
  float* bufA     = ws + 4 * (size_t)N;                     // N*128: hproj1, later hproj2+agg2
  float* bufB     = bufA + (size_t)N * GCN_HID_DIM;          // N*128: agg1 -> h1 (in place)
  float* hproj2   = bufA;                                    // N*64 (hproj1 dead by then)
  float* agg2     = bufA + (size_t)N * GCN_OUT_DIM;          // N*64

  // Degrees + normalization
  hipMemsetAsync(deg_out, 0, 2 * (size_t)N * sizeof(float), stream);
  gcn_degree_kernel<<<(E + 255) / 256, 256, 0, stream>>>(src, dst, deg_out, deg_in, E);
  gcn_norm_kernel<<<(N + 255) / 256, 256, 0, stream>>>(deg_out, deg_in, norm_out, norm_in, N);

  // ---- Layer 1: hproj1 = (feat * norm_out) @ W1 ----
  gcn_gemm_scaled_wmma<GCN_IN_DIM, GCN_HID_DIM>
      <<<N / 16, (GCN_HID_DIM / 16) * 32, 0, stream>>>(feat, norm_out, W1, bufA);
  hipMemsetAsync(bufB, 0, (size_t)N * GCN_HID_DIM * sizeof(float), stream);
  {
    const unsigned work = (unsigned)E * GCN_HID_DIM;
    gcn_aggregate_kernel<GCN_HID_DIM>
        <<<(work + 255) / 256, 256, 0, stream>>>(bufA, src, dst, bufB, E);
  }
  {
    const unsigned work = (unsigned)N * GCN_HID_DIM;
    gcn_finalize_kernel<GCN_HID_DIM>
        <<<(work + 255) / 256, 256, 0, stream>>>(bufB, norm_in, b1, bufB, N);  // h1 in-place
  }

  // ---- Layer 2: hproj2 = (h1 * norm_out) @ W2 ----
  gcn_gemm_scaled_wmma<GCN_HID_DIM, GCN_OUT_DIM>
      <<<N / 16, (GCN_OUT_DIM / 16) * 32, 0, stream>>>(bufB, norm_out, W2, hproj2);
  hipMemsetAsync(agg2, 0, (size_t)N * GCN_OUT_DIM * sizeof(float), stream);
  {
    const unsigned work = (unsigned)E * GCN_OUT_DIM;
    gcn_aggregate_kernel<GCN_OUT_DIM>
        <<<(work + 255) / 256, 256, 0, stream>>>(hproj2, src, dst, agg2, E);
  }
  {
    const unsigned work = (unsigned)N * GCN_OUT_DIM;
    gcn_finalize_kernel<GCN_OUT_DIM>
        <<<(work + 255) / 256, 256, 0, stream>>>(agg2, norm_in, b2, out, N);
  }
}